// NeighbourDotAttention_79680233275439
// MI455X (gfx1250) — compile-verified
//
#include <hip/hip_runtime.h>
#include <stdint.h>

// CDNA5 / gfx1250: segment-sum of gathered rows via the Tensor Data Mover
// (gather mode), double-buffered in LDS, one wave32 per destination node.
//
// The reference's softmax is over a singleton axis -> attn == 1.0 exactly, so
// the whole attention/embedding path is dead code and the output is exactly
//   out[n,:] = sum_{e : seg_ids[e]==n} source[src_idx[e], :]

typedef unsigned int u32x4 __attribute__((ext_vector_type(4)));
typedef int          i32x4 __attribute__((ext_vector_type(4)));
typedef int          i32x8 __attribute__((ext_vector_type(8)));

#define ROW_SIZE 64   // fp32 elements per row (256 bytes)
#define GROWS    8    // TDM gather-mode rows per op (32-bit index mode)

static __device__ __forceinline__ int imin(int a, int b) { return a < b ? a : b; }

// Low 32 bits of a generic (flat) LDS address are the wave-relative LDS byte
// offset (aperture mapping).
static __device__ __forceinline__ uint32_t lds_offset(const void* p) {
  return (uint32_t)(uintptr_t)p;
}

// Issue one TDM gather: cnt (<=8) rows of 64 fp32 from source[idx[j]*64 .. +63]
// into LDS at lds_addr. Tracked by TENSORcnt.
static __device__ __forceinline__ void tdm_gather8(
    const float* __restrict__ source, const int* __restrict__ src_idx,
    int e0, int cnt, uint32_t lds_addr, uint32_t n_nodes, int lane)
{
  int v = 0;
  if (lane < cnt) v = src_idx[e0 + lane];        // lanes 0..cnt-1 hold row indices
  const int i0 = __builtin_amdgcn_readlane(v, 0);
  const int i1 = __builtin_amdgcn_readlane(v, 1);
  const int i2 = __builtin_amdgcn_readlane(v, 2);
  const int i3 = __builtin_amdgcn_readlane(v, 3);
  const int i4 = __builtin_amdgcn_readlane(v, 4);
  const int i5 = __builtin_amdgcn_readlane(v, 5);
  const int i6 = __builtin_amdgcn_readlane(v, 6);
  const int i7 = __builtin_amdgcn_readlane(v, 7);

  const uint64_t ga = (uint64_t)(uintptr_t)source;

  // ---- D# group 0 (128b): count=1, gather_index_size=32b, gather_mode=1,
  //      lds_addr, global_addr(57b), type=2 ("image").
  u32x4 g0;
  g0.x = 1u | (1u << 30) | (1u << 31);
  g0.y = lds_addr;
  g0.z = (uint32_t)ga;
  g0.w = ((uint32_t)(ga >> 32) & 0x01FFFFFFu) | (2u << 30);

  // ---- D# group 1 (256b)
  i32x8 g1;
  g1[0] = (int)(2u << 16);                             // wg_mask=0, data_size=2 (4B)
  g1[1] = (int)((uint32_t)ROW_SIZE << 16);             // tensor_dim0[15:0]=64
  g1[2] = (int)((n_nodes & 0xFFFFu) << 16);            // tensor_dim0 hi=0 | tensor_dim1 lo
  g1[3] = (int)(((n_nodes >> 16) & 0xFFFFu)            // tensor_dim1 hi
                | ((uint32_t)ROW_SIZE << 16));         // tile_dim0 = 64
  g1[4] = cnt & 0xFFFF;                                // tile_dim1 = #valid indices
  g1[5] = ROW_SIZE;                                    // tensor_dim0_stride lo32 = 64
  g1[6] = 0;                                           // stride hi / dim1_stride (ignored)
  g1[7] = 0;

  // ---- D# groups 2/3: 32-bit gather row indices (up to 8 rows)
  i32x4 g2 = { i0, i1, i2, i3 };
  i32x4 g3 = { i4, i5, i6, i7 };

  // This toolchain exposes the 6-arg builtin (clang-23 / therock-10.0 form):
  // (uint32x4, int32x8, int32x4, int32x4, int32x8, i32 cpol). The D# only has
  // four SGPR groups; the trailing int32x8 is zero-filled.
  i32x8 gz = { 0, 0, 0, 0, 0, 0, 0, 0 };
  __builtin_amdgcn_tensor_load_to_lds(g0, g1, g2, g3, gz, 0);
}

__global__ __launch_bounds__(32)
void NeighbourDotAttention_79680233275439_kernel(
    const float* __restrict__ source,
    const int*   __restrict__ src_idx,
    const int*   __restrict__ seg_ids,
    float*       __restrict__ out,
    int n_nodes, int n_edges)
{
  __shared__ float buf[2][GROWS * ROW_SIZE];   // 2 x 2KB double buffer

  const int node = (int)blockIdx.x;
  const int lane = (int)threadIdx.x;

  // seg_ids is sorted: binary search this node's edge range (uniform per wave).
  int lo = 0, hi = n_edges;
  while (lo < hi) { int m = (lo + hi) >> 1; if (seg_ids[m] <  node) lo = m + 1; else hi = m; }
  const int start = lo;
  hi = n_edges;
  while (lo < hi) { int m = (lo + hi) >> 1; if (seg_ids[m] <= node) lo = m + 1; else hi = m; }
  const int end = lo;

  const int nedge   = end - start;
  const int nchunks = (nedge + GROWS - 1) / GROWS;

  float acc0 = 0.0f, acc1 = 0.0f;

  if (nchunks > 0) {
    tdm_gather8(source, src_idx, start, imin(nedge, GROWS),
                lds_offset(&buf[0][0]), (uint32_t)n_nodes, lane);
  }

  for (int c = 0; c < nchunks; ++c) {
    const int cnt = imin(nedge - c * GROWS, GROWS);

    if (c + 1 < nchunks) {
      // Stream the index list ahead and issue the next gather while we reduce
      // the current buffer. TDM ops from one wave complete in order, so
      // tensorcnt<=1 means chunk c has landed in LDS.
      __builtin_prefetch(src_idx + start + (c + 2) * GROWS, 0, 1);
      tdm_gather8(source, src_idx, start + (c + 1) * GROWS,
                  imin(nedge - (c + 1) * GROWS, GROWS),
                  lds_offset(&buf[(c + 1) & 1][0]), (uint32_t)n_nodes, lane);
      __builtin_amdgcn_s_wait_tensorcnt(1);
    } else {
      __builtin_amdgcn_s_wait_tensorcnt(0);
    }
    asm volatile("" ::: "memory");   // LDS was written by the TDM, force reloads

    const float* b = &buf[c & 1][0];
    #pragma unroll 4
    for (int r = 0; r < cnt; ++r) {
      acc0 += b[r * ROW_SIZE + lane];
      acc1 += b[r * ROW_SIZE + lane + 32];
    }
  }

  // Every node writes (zero-edge nodes produce zeros, matching segment_sum).
  out[(size_t)node * ROW_SIZE + lane]      = acc0;
  out[(size_t)node * ROW_SIZE + lane + 32] = acc1;
}

extern "C" void kernel_launch(void* const* d_in, const int* in_sizes, int n_in,
                              void* d_out, int out_size, void* d_ws, size_t ws_size,
                              hipStream_t stream) {
  // setup_inputs order: source, target, src_idx, seg_ids, W_emb, b_emb,
  //                     w_loc, b_loc, w_nb, b_nb
  // Everything except source/src_idx/seg_ids is dead code (softmax over a
  // singleton axis == 1.0 exactly).
  const float* source  = (const float*)d_in[0];
  const int*   src_idx = (const int*)d_in[2];
  const int*   seg_ids = (const int*)d_in[3];
  float*       out     = (float*)d_out;

  const int n_edges = in_sizes[2];
  const int n_nodes = out_size / ROW_SIZE;

  dim3 grid((unsigned)n_nodes), block(32);
  hipLaunchKernelGGL(NeighbourDotAttention_79680233275439_kernel,
                     grid, block, 0, stream,
                     source, src_idx, seg_ids, out, n_nodes, n_edges);
}